// fofe_tricontext_62569083568228
// MI455X (gfx1250) — compile-verified
//
#include <hip/hip_runtime.h>
#include <math.h>

// ---------------------------------------------------------------------------
// FOFE tri-context encoder, restructured:
//   L[p] = sum_{j<=p} a^(p-j) x[j]   (prefix FOFE)   = T_low  @ X
//   R[p] = sum_{j>=p} a^(j-p) x[j]   (suffix FOFE)   = T_low^T @ X
// Per candidate (s, e):
//   k=0 left-excl  = L[s-1]            (0 if s==0)
//   k=1 left-incl  = L[e]
//   k=2 candidate  = L[e] - a^(e-s+1) * L[s-1]
//   k=3 right-incl = R[s]
//   k=4 right-excl = R[e+1]            (0 if e==DOC-1)
// Phase 1: two triangular GEMMs via V_WMMA_F32_16X16X4_F32 (fp32 in/out),
//          wave-uniform (readfirstlane) scalar loop control, branch-free A
//          generation from a zero-padded LDS alpha table, triangular K-range
//          skipping (all-zero A tiles never issued).
// Phase 2: bandwidth-bound gather/axpy writing the 488 MB output
//          (store floor: 488 MB / 23.3 TB/s ~ 21 us).
// ---------------------------------------------------------------------------

#define DOCL   256
#define EDIM   768
#define NBATCH 8
#define MAXC   16
#define NCAND  3976                 // 240*16 + (16+15+...+1)
#define LOG2_ALPHA -0.15200309344504997   // log2(0.9), double

#define NT_M (DOCL / 16)            // 16 M-tiles
#define NT_N (EDIM / 16)            // 48 N-tiles
#define WAVES_PER_BLOCK 4
#define TILES_TOTAL (2 * NBATCH * NT_M * NT_N)          // 12288
#define SCAN_BLOCKS (TILES_TOTAL / WAVES_PER_BLOCK)     // 3072

typedef __attribute__((ext_vector_type(2))) float v2f;
typedef __attribute__((ext_vector_type(8))) float v8f;

// ---------------------------------------------------------------------------
// Phase 1: L = T_low @ X, R = T_low^T @ X  (per batch), via fp32 WMMA.
// One wave per 16x16 output tile. ISA operand layouts (CDNA5 ISA):
//   A 16x4 f32 : lane l, vgpr i -> M = l&15, K = (l>>4)*2 + i
//   B 4x16 f32 : lane l, vgpr i -> N = l&15, K = (l>>4)*2 + i
//   C/D 16x16  : lane l, vgpr r -> N = l&15, M = r + (l>>4)*8
// A values come from a 512-entry LDS table: [0,255]=0, [256,511]=alpha^(i-256),
// indexed by 256 + sgn*(p - kk) -> unconditional ds_load_b32, no predication.
// Tile coordinates are forced into SGPRs via readfirstlane so the K loop is
// scalar-controlled (s_cmp/s_cbranch) and unrollable.
// ---------------------------------------------------------------------------
__global__ __launch_bounds__(32 * WAVES_PER_BLOCK)
void fofe_scan_wmma_kernel(const float* __restrict__ x,
                           float* __restrict__ Lbuf,
                           float* __restrict__ Rbuf)
{
    __shared__ float apow2[2 * DOCL];
    for (int k = threadIdx.x; k < 2 * DOCL; k += 32 * WAVES_PER_BLOCK)
        apow2[k] = (k < DOCL) ? 0.0f
                              : (float)exp2((double)(k - DOCL) * LOG2_ALPHA);
    __syncthreads();

    // Wave-uniform tile coordinates (identical across the wave's 32 lanes).
    int w = __builtin_amdgcn_readfirstlane(
                (int)(blockIdx.x * WAVES_PER_BLOCK) + (int)(threadIdx.x >> 5));
    int lane = threadIdx.x & 31;

    int nt = w % NT_N;   w /= NT_N;
    int mt = w % NT_M;   w /= NT_M;
    int b  = w % NBATCH; w /= NBATCH;
    int upper = w;       // 0 -> L (alpha^(p-j), j<=p), 1 -> R (alpha^(j-p), j>=p)

    const int row   = lane & 15;   // M index (A) and N index (B/D)
    const int khalf = lane >> 4;   // selects K pair
    const int m0 = mt * 16;
    const int n0 = nt * 16;
    const int p  = m0 + row;       // output document position (A row)

    // Triangular structure: A is identically zero outside these K ranges.
    const int sgn  = upper ? -1 : 1;          // table idx = 256 + sgn*(p - kk)
    const int kbeg = upper ? m0 : 0;          // uniform (SGPR)
    const int kend = upper ? DOCL : (m0 + 16);// uniform (SGPR)

    const float* __restrict__ xcol =
        x + (size_t)b * DOCL * EDIM + (n0 + row);

    v8f acc = {};
    #pragma unroll 4
    for (int k0 = kbeg; k0 < kend; k0 += 4) {
        const int kk0 = k0 + khalf * 2;
        v2f av, bv;
        av[0] = apow2[DOCL + sgn * (p - kk0)];
        av[1] = apow2[DOCL + sgn * (p - kk0 - 1)];
        bv[0] = xcol[(size_t)kk0 * EDIM];
        bv[1] = xcol[(size_t)(kk0 + 1) * EDIM];
        acc = __builtin_amdgcn_wmma_f32_16x16x4_f32(
                false, av, false, bv, (short)0, acc, false, false);
    }

    float* __restrict__ ob = (upper ? Rbuf : Lbuf) + (size_t)b * DOCL * EDIM;
    #pragma unroll
    for (int r = 0; r < 8; ++r) {
        int pr = m0 + khalf * 8 + r;
        ob[(size_t)pr * EDIM + (n0 + row)] = acc[r];
    }
}

// ---------------------------------------------------------------------------
// Candidate index -> (start, end). Candidates are ordered by start then span:
// starts 0..239 have 16 spans each (3840 candidates), then a triangular tail
// 16,15,...,1 for starts 240..255.
// ---------------------------------------------------------------------------
__device__ __forceinline__ void cand_to_se(int c, int& s, int& e)
{
    const int dense = (DOCL - MAXC) * MAXC;   // 3840
    if (c < dense) {
        s = c >> 4;
        e = s + (c & 15);
    } else {
        int cp = c - dense;
        int k = 0, base = 0;
        while (cp >= base + (MAXC - k)) { base += MAXC - k; ++k; }
        s = (DOCL - MAXC) + k;
        e = s + (cp - base);
    }
}

// ---------------------------------------------------------------------------
// Phase 2: assemble outputs. One 192-thread block (6 wave32) per (b, cand);
// each thread owns one float4 column group (EDIM/4 = 192). Loads up to 4 L/R
// rows from L2-resident scratch, writes 5 contiguous 3KB rows.
// ---------------------------------------------------------------------------
__global__ __launch_bounds__(EDIM / 4)
void fofe_gather_kernel(const float* __restrict__ Lbuf,
                        const float* __restrict__ Rbuf,
                        const int*   __restrict__ mask,
                        float* __restrict__ out_codes,
                        float* __restrict__ out_pos,
                        float* __restrict__ out_pad)
{
    const int c = blockIdx.x;
    const int b = blockIdx.y;
    int s, e;
    cand_to_se(c, s, e);

    // a = alpha^(e-s+1), span+1 <= 16: short uniform double product (float-exact)
    double ad = 1.0;
    for (int j = 0; j <= e - s; ++j) ad *= 0.9;
    const float a = (float)ad;

    const int t  = threadIdx.x;       // float4 column group
    const int NV = EDIM / 4;          // 192 float4 per row

    const float4* __restrict__ Lb = (const float4*)(Lbuf + (size_t)b * DOCL * EDIM);
    const float4* __restrict__ Rb = (const float4*)(Rbuf + (size_t)b * DOCL * EDIM);

    const float4 z  = make_float4(0.f, 0.f, 0.f, 0.f);
    const float4 l1 = (s > 0)        ? Lb[(size_t)(s - 1) * NV + t] : z;  // left-excl
    const float4 l2 =                  Lb[(size_t)e       * NV + t];      // left-incl
    const float4 r4 =                  Rb[(size_t)s       * NV + t];      // right-incl
    const float4 r3 = (e < DOCL - 1) ? Rb[(size_t)(e + 1) * NV + t] : z;  // right-excl

    float4 c0;                                                            // candidate
    c0.x = fmaf(-a, l1.x, l2.x);
    c0.y = fmaf(-a, l1.y, l2.y);
    c0.z = fmaf(-a, l1.z, l2.z);
    c0.w = fmaf(-a, l1.w, l2.w);

    const size_t base = ((size_t)b * NCAND + c) * 5 * EDIM;
    float4* __restrict__ o = (float4*)(out_codes + base);
    o[0 * NV + t] = l1;   // buf1: left ctx excl cand
    o[1 * NV + t] = l2;   // buf2: left ctx incl cand
    o[2 * NV + t] = c0;   // buf0: candidate ctx
    o[3 * NV + t] = r4;   // buf4: right ctx incl cand
    o[4 * NV + t] = r3;   // buf3: right ctx excl cand

    if (t == 0) {
        const size_t bc = (size_t)b * NCAND + c;
        out_pos[bc * 2 + 0] = (float)s;
        out_pos[bc * 2 + 1] = (float)e;
        int any = 0;
        for (int j = s; j <= e; ++j) any |= mask[b * DOCL + j];
        out_pad[bc] = any ? 1.0f : 0.0f;
    }
}

// ---------------------------------------------------------------------------
extern "C" void kernel_launch(void* const* d_in, const int* in_sizes, int n_in,
                              void* d_out, int out_size, void* d_ws, size_t ws_size,
                              hipStream_t stream)
{
    (void)in_sizes; (void)n_in; (void)out_size; (void)ws_size;

    const float* x    = (const float*)d_in[0];   // [8, 256, 768] f32
    const int*   mask = (const int*)  d_in[1];   // [8, 256] i32

    // Scratch: L and R scans, each [8, 256, 768] f32 (6.29 MB) -> 12.6 MB.
    float* Lbuf = (float*)d_ws;
    float* Rbuf = Lbuf + (size_t)NBATCH * DOCL * EDIM;

    // Output layout (flat, return order): fofe_codes | cands_pos | padded_cands
    float* out_codes = (float*)d_out;
    float* out_pos   = out_codes + (size_t)NBATCH * NCAND * 5 * EDIM;
    float* out_pad   = out_pos   + (size_t)NBATCH * NCAND * 2;

    fofe_scan_wmma_kernel<<<dim3(SCAN_BLOCKS), dim3(32 * WAVES_PER_BLOCK), 0, stream>>>(
        x, Lbuf, Rbuf);

    fofe_gather_kernel<<<dim3(NCAND, NBATCH), dim3(EDIM / 4), 0, stream>>>(
        Lbuf, Rbuf, mask, out_codes, out_pos, out_pad);
}